// EGNNLayer_14843406975721
// MI455X (gfx1250) — compile-verified
//
#include <hip/hip_runtime.h>
#include <math.h>

typedef __attribute__((ext_vector_type(2))) float v2f;
typedef __attribute__((ext_vector_type(8))) float v8f;

#define N_NODES 50000
#define N_EDGES 800000
#define EPB 64  // edges per block (4 M-tiles of 16)

// D = A(16x4) * B(4x16) + C(16x16), all fp32 (V_WMMA_F32_16X16X4_F32)
__device__ __forceinline__ v8f wmma4(v2f a, v2f b, v8f c) {
  return __builtin_amdgcn_wmma_f32_16x16x4_f32(false, a, false, b, (short)0, c,
                                               false, false);
}

__device__ __forceinline__ float silu_f(float v) {
  return v / (1.0f + __expf(-v));
}

// ---------------------------------------------------------------------------
// Init: zero agg workspace, seed coords_out with coords (atomics add onto it)
// ---------------------------------------------------------------------------
__global__ void egnn_init_kernel(const float* __restrict__ coords,
                                 float* __restrict__ agg,
                                 float* __restrict__ coords_out) {
  int i = blockIdx.x * blockDim.x + threadIdx.x;
  if (i < N_NODES * 128) agg[i] = 0.0f;
  if (i < N_NODES * 3) coords_out[i] = coords[i];
}

// ---------------------------------------------------------------------------
// Edge kernel: 64 edges per block, 8 waves x 16 output features, 4 M-tiles
// per wave so each B fragment is loaded once and fed to 4 WMMAs.
// Stage1: h1 = silu([x_r|x_c|ea] @ We1 + be1 + dist*We1[288,:])   (K=288)
// Stage2: ef = h1 @ We2 + be2 ; atomicAdd into agg[row]           (K=128)
// Stage3: w  = silu(ef @ Wc1 + bc1) @ Wc2  (per-edge scalar)      (K=128)
// Stage4: coords_out[row] += (diff/(sqrt(dist)+eps)) * w
// ---------------------------------------------------------------------------
__global__ __launch_bounds__(256) void egnn_edge_kernel(
    const float* __restrict__ x, const float* __restrict__ coords,
    const float* __restrict__ edge_attr, const int* __restrict__ erow,
    const int* __restrict__ ecol, const float* __restrict__ We1,
    const float* __restrict__ be1, const float* __restrict__ We2,
    const float* __restrict__ be2, const float* __restrict__ Wc1,
    const float* __restrict__ bc1, const float* __restrict__ Wc2,
    float* __restrict__ agg, float* __restrict__ coords_out) {
  // sA: [64][288] stride 292 (stages 0-1), then reused as sE [64][128] s132
  // sH: [64][128] stride 132
  __shared__ float smem[EPB * 292 + EPB * 132];
  float* sA = smem;
  float* sH = smem + EPB * 292;
  float* sE = smem;  // overlays sA (dead after stage 1)

  __shared__ int sRow[EPB], sCol[EPB];
  __shared__ float sDiff[EPB * 3], sDist[EPB], sW[EPB];

  const int tid = threadIdx.x;
  const int lane = tid & 31;
  const int wv = tid >> 5;   // wave id 0..7 -> output feature slice
  const int hi = lane >> 4;  // half-wave select
  const int lm = lane & 15;
  const int n0 = wv * 16;
  const int ebase = blockIdx.x * EPB;

  if (tid < EPB) {
    int r = erow[ebase + tid];
    int c = ecol[ebase + tid];
    sRow[tid] = r;
    sCol[tid] = c;
    float d0 = coords[r * 3 + 0] - coords[c * 3 + 0];
    float d1 = coords[r * 3 + 1] - coords[c * 3 + 1];
    float d2 = coords[r * 3 + 2] - coords[c * 3 + 2];
    sDiff[tid * 3 + 0] = d0;
    sDiff[tid * 3 + 1] = d1;
    sDiff[tid * 3 + 2] = d2;
    sDist[tid] = d0 * d0 + d1 * d1 + d2 * d2;
    sW[tid] = 0.0f;
  }
  __syncthreads();

  // Gather A tile: cols 0-127 x[row], 128-255 x[col], 256-287 edge_attr
  for (int idx = tid; idx < EPB * 288; idx += 256) {
    int e = idx / 288;
    int c = idx - e * 288;
    float v;
    if (c < 128)
      v = x[sRow[e] * 128 + c];
    else if (c < 256)
      v = x[sCol[e] * 128 + (c - 128)];
    else
      v = edge_attr[(ebase + e) * 32 + (c - 256)];
    sA[e * 292 + c] = v;
  }
  __syncthreads();

  // ---- Stage 1: K=288 over We1; dist column (row 288) as rank-1 epilogue
  {
    v8f acc[4] = {{}, {}, {}, {}};
    for (int k0 = 0; k0 < 288; k0 += 4) {
      int kk = k0 + 2 * hi;
      v2f b;
      b.x = We1[kk * 128 + n0 + lm];
      b.y = We1[(kk + 1) * 128 + n0 + lm];
#pragma unroll
      for (int t = 0; t < 4; ++t) {
        v2f a;
        a.x = sA[(t * 16 + lm) * 292 + kk];
        a.y = sA[(t * 16 + lm) * 292 + kk + 1];
        acc[t] = wmma4(a, b, acc[t]);
      }
    }
    float bias = be1[n0 + lm];
    float wlast = We1[288 * 128 + n0 + lm];
#pragma unroll
    for (int t = 0; t < 4; ++t) {
#pragma unroll
      for (int v = 0; v < 8; ++v) {
        int M = t * 16 + v + 8 * hi;
        float pre = acc[t][v] + bias + sDist[M] * wlast;
        sH[M * 132 + n0 + lm] = silu_f(pre);
      }
    }
  }
  __syncthreads();  // all reads of sA done; sE may now overlay it

  // ---- Stage 2: edge_feat = h1 @ We2 + be2 ; scatter-add into agg[row]
  {
    v8f acc[4] = {{}, {}, {}, {}};
    for (int k0 = 0; k0 < 128; k0 += 4) {
      int kk = k0 + 2 * hi;
      v2f b;
      b.x = We2[kk * 128 + n0 + lm];
      b.y = We2[(kk + 1) * 128 + n0 + lm];
#pragma unroll
      for (int t = 0; t < 4; ++t) {
        v2f a;
        a.x = sH[(t * 16 + lm) * 132 + kk];
        a.y = sH[(t * 16 + lm) * 132 + kk + 1];
        acc[t] = wmma4(a, b, acc[t]);
      }
    }
    float bias = be2[n0 + lm];
#pragma unroll
    for (int t = 0; t < 4; ++t) {
#pragma unroll
      for (int v = 0; v < 8; ++v) {
        int M = t * 16 + v + 8 * hi;
        float ef = acc[t][v] + bias;
        sE[M * 132 + n0 + lm] = ef;
        atomicAdd(&agg[sRow[M] * 128 + n0 + lm], ef);
      }
    }
  }
  __syncthreads();

  // ---- Stage 3: per-edge gate w = silu(ef @ Wc1 + bc1) . Wc2
  {
    v8f acc[4] = {{}, {}, {}, {}};
    for (int k0 = 0; k0 < 128; k0 += 4) {
      int kk = k0 + 2 * hi;
      v2f b;
      b.x = Wc1[kk * 128 + n0 + lm];
      b.y = Wc1[(kk + 1) * 128 + n0 + lm];
#pragma unroll
      for (int t = 0; t < 4; ++t) {
        v2f a;
        a.x = sE[(t * 16 + lm) * 132 + kk];
        a.y = sE[(t * 16 + lm) * 132 + kk + 1];
        acc[t] = wmma4(a, b, acc[t]);
      }
    }
    float bias = bc1[n0 + lm];
    float wc2 = Wc2[n0 + lm];
#pragma unroll
    for (int t = 0; t < 4; ++t) {
#pragma unroll
      for (int v = 0; v < 8; ++v) {
        int M = t * 16 + v + 8 * hi;
        float s = silu_f(acc[t][v] + bias);
        atomicAdd(&sW[M], s * wc2);  // ds_add_f32 reduction over feature dim
      }
    }
  }
  __syncthreads();

  // ---- Stage 4: coordinate scatter (64 edges x 3 dims = 192 threads)
  if (tid < EPB * 3) {
    int e = tid / 3;
    int d = tid - e * 3;
    float cd = sDiff[tid] / (sqrtf(sDist[e]) + 1e-8f);
    atomicAdd(&coords_out[sRow[e] * 3 + d], cd * sW[e]);
  }
}

// ---------------------------------------------------------------------------
// Node kernel: tile of 16 nodes, x_new = silu([x|agg] @ Wn1 + bn1) @ Wn2 + bn2
// ---------------------------------------------------------------------------
__global__ __launch_bounds__(256) void egnn_node_kernel(
    const float* __restrict__ x, const float* __restrict__ agg,
    const float* __restrict__ Wn1, const float* __restrict__ bn1,
    const float* __restrict__ Wn2, const float* __restrict__ bn2,
    float* __restrict__ x_out) {
  __shared__ float sA[16 * 260];  // [x|agg] [16][256], stride 260
  __shared__ float sH[16 * 132];

  const int tid = threadIdx.x;
  const int lane = tid & 31;
  const int wv = tid >> 5;
  const int hi = lane >> 4;
  const int lm = lane & 15;
  const int n0 = wv * 16;
  const int nb = blockIdx.x * 16;

  for (int idx = tid; idx < 16 * 256; idx += 256) {
    int m = idx >> 8;
    int c = idx & 255;
    float v = (c < 128) ? x[(nb + m) * 128 + c] : agg[(nb + m) * 128 + (c - 128)];
    sA[m * 260 + c] = v;
  }
  __syncthreads();

  v8f acc1 = {};
  for (int k0 = 0; k0 < 256; k0 += 4) {
    int kk = k0 + 2 * hi;
    v2f a, b;
    a.x = sA[lm * 260 + kk];
    a.y = sA[lm * 260 + kk + 1];
    b.x = Wn1[kk * 128 + n0 + lm];
    b.y = Wn1[(kk + 1) * 128 + n0 + lm];
    acc1 = wmma4(a, b, acc1);
  }
  {
    float bias = bn1[n0 + lm];
#pragma unroll
    for (int v = 0; v < 8; ++v) {
      int M = v + 8 * hi;
      sH[M * 132 + n0 + lm] = silu_f(acc1[v] + bias);
    }
  }
  __syncthreads();

  v8f acc2 = {};
  for (int k0 = 0; k0 < 128; k0 += 4) {
    int kk = k0 + 2 * hi;
    v2f a, b;
    a.x = sH[lm * 132 + kk];
    a.y = sH[lm * 132 + kk + 1];
    b.x = Wn2[kk * 128 + n0 + lm];
    b.y = Wn2[(kk + 1) * 128 + n0 + lm];
    acc2 = wmma4(a, b, acc2);
  }
  {
    float bias = bn2[n0 + lm];
#pragma unroll
    for (int v = 0; v < 8; ++v) {
      int M = v + 8 * hi;
      x_out[(nb + M) * 128 + n0 + lm] = acc2[v] + bias;
    }
  }
}

// ---------------------------------------------------------------------------
extern "C" void kernel_launch(void* const* d_in, const int* in_sizes, int n_in,
                              void* d_out, int out_size, void* d_ws,
                              size_t ws_size, hipStream_t stream) {
  const float* x = (const float*)d_in[0];
  const float* coords = (const float*)d_in[1];
  const float* edge_attr = (const float*)d_in[2];
  const int* edge_index = (const int*)d_in[3];
  const float* We1 = (const float*)d_in[4];
  const float* be1 = (const float*)d_in[5];
  const float* We2 = (const float*)d_in[6];
  const float* be2 = (const float*)d_in[7];
  const float* Wc1 = (const float*)d_in[8];
  const float* bc1 = (const float*)d_in[9];
  const float* Wc2 = (const float*)d_in[10];
  const float* Wn1 = (const float*)d_in[11];
  const float* bn1 = (const float*)d_in[12];
  const float* Wn2 = (const float*)d_in[13];
  const float* bn2 = (const float*)d_in[14];

  float* x_out = (float*)d_out;                       // [N,128]
  float* coords_out = x_out + (size_t)N_NODES * 128;  // [N,3]
  float* agg = (float*)d_ws;                          // [N,128] scratch

  const int* erow = edge_index;
  const int* ecol = edge_index + N_EDGES;

  egnn_init_kernel<<<(N_NODES * 128 + 255) / 256, 256, 0, stream>>>(
      coords, agg, coords_out);
  egnn_edge_kernel<<<N_EDGES / EPB, 256, 0, stream>>>(
      x, coords, edge_attr, erow, ecol, We1, be1, We2, be2, Wc1, bc1, Wc2, agg,
      coords_out);
  egnn_node_kernel<<<N_NODES / 16, 256, 0, stream>>>(x, agg, Wn1, bn1, Wn2, bn2,
                                                     x_out);
}